// ODEAttnEncoder_36661840839113
// MI455X (gfx1250) — compile-verified
//
#include <hip/hip_runtime.h>
#include <math.h>

typedef __attribute__((ext_vector_type(2))) float v2f;
typedef __attribute__((ext_vector_type(4))) float v4f;
typedef __attribute__((ext_vector_type(8))) float v8f;

#define B_    32
#define S_    64
#define MTOT  2048       // B*S
#define FEAT_ 128
#define DM    256
#define INNER_ 1024
#define HID_  512
#define SDIM_ 6
#define KODE  272        // 265 padded to multiple of 16
#define KC    16         // K-chunk staged in LDS per double-buffer phase
#define LSTR  20         // LDS row stride (floats): 16B-aligned rows, bank-spread

static inline int cdiv(int a, int b) { return (a + b - 1) / b; }

// ---------------------------------------------------------------------------
// fp32 WMMA GEMM with LDS double-buffering:
//   C[M,N] = act(A[M,K] @ W[N,K]^T + bias) (+ resid)
// Block tile 64x128, 8 waves (wave32). Per K-chunk of 16:
//   - 256 threads cooperatively stage A(64x16) and B(128x16) tiles into LDS
//     with coalesced b128 global loads (issued one chunk AHEAD of compute,
//     so global latency hides behind 16 WMMAs of the current chunk),
//   - each wave (wm 0..3, wn 0..1) runs 4 k-steps x 4 column strips of
//     V_WMMA_F32_16X16X4_F32 fed by ds_load_b64 from LDS.
// ISA lane layouts (CDNA5 ISA):
//   A 16x4 : lane l, vgpr j -> A[m0+(l&15), k + 2*(l>>4) + j]
//   B 4x16 : lane l, vgpr j -> B[k + 2*(l>>4) + j, n0+(l&15)]  (B = W^T)
//   C 16x16: lane l, vgpr v -> C[m0 + v + 8*(l>>4), n0+(l&15)]
// GUARD=0: N multiple of 128 (no masking anywhere).
// GUARD=1: out-of-range W rows zero-masked at staging (used only for N=16).
// act: 0=none, 1=relu, 2=elu.  M, K multiples of 64 / 16 (true for all calls).
// ---------------------------------------------------------------------------
template <int GUARD>
__global__ __launch_bounds__(256) void k_gemm(
    const float* __restrict__ A, int lda,
    const float* __restrict__ W, int ldw,
    const float* __restrict__ bias, int bias_n,
    float* __restrict__ C, int ldc,
    const float* __restrict__ resid, int ldr,
    int N, int K, int act)
{
  __shared__ float As[2][64][LSTR];
  __shared__ float Bs[2][128][LSTR];

  const int t    = threadIdx.x;
  const int lane = t & 31;
  const int wave = t >> 5;
  const int wm = wave >> 1, wn = wave & 1;
  const int m0 = blockIdx.y * 64 + wm * 16;
  const int n0 = blockIdx.x * 128 + wn * 64;
  const int l15 = lane & 15, half = lane >> 4;

  // ---- staging assignments (per thread): 1 A float4 slot, 2 B float4 slots
  const int ar = t >> 2, ac = t & 3;           // A tile row / float4-col
  const int br = t >> 2, bc = t & 3;           // B slots: rows br and br+64
  const float* aSrc = A + (size_t)(blockIdx.y * 64 + ar) * lda + 4 * ac;

  int wr0 = blockIdx.x * 128 + br;
  int wr1 = wr0 + 64;
  float sB0 = 1.f, sB1 = 1.f;
  if (GUARD) {
    sB0 = (wr0 < N) ? 1.f : 0.f; if (wr0 >= N) wr0 = 0;
    sB1 = (wr1 < N) ? 1.f : 0.f; if (wr1 >= N) wr1 = 0;
  }
  const float* bSrc0 = W + (size_t)wr0 * ldw + 4 * bc;
  const float* bSrc1 = W + (size_t)wr1 * ldw + 4 * bc;

  v8f acc0 = {0.f,0.f,0.f,0.f,0.f,0.f,0.f,0.f};
  v8f acc1 = acc0, acc2 = acc0, acc3 = acc0;

  auto compute = [&](const float (*Ab)[LSTR], const float (*Bb)[LSTR]) {
#pragma unroll
    for (int kk = 0; kk < KC; kk += 4) {
      v2f a  = *(const v2f*)&Ab[wm * 16 + l15][kk + 2 * half];
      v2f b0 = *(const v2f*)&Bb[wn * 64 +  0 + l15][kk + 2 * half];
      v2f b1 = *(const v2f*)&Bb[wn * 64 + 16 + l15][kk + 2 * half];
      v2f b2 = *(const v2f*)&Bb[wn * 64 + 32 + l15][kk + 2 * half];
      v2f b3 = *(const v2f*)&Bb[wn * 64 + 48 + l15][kk + 2 * half];
      acc0 = __builtin_amdgcn_wmma_f32_16x16x4_f32(false, a, false, b0,
                                                   (short)0, acc0, false, false);
      acc1 = __builtin_amdgcn_wmma_f32_16x16x4_f32(false, a, false, b1,
                                                   (short)0, acc1, false, false);
      acc2 = __builtin_amdgcn_wmma_f32_16x16x4_f32(false, a, false, b2,
                                                   (short)0, acc2, false, false);
      acc3 = __builtin_amdgcn_wmma_f32_16x16x4_f32(false, a, false, b3,
                                                   (short)0, acc3, false, false);
    }
  };

  // ---- prologue: stage chunk 0 into buffer 0
  {
    v4f ga  = *(const v4f*)(aSrc);
    v4f gb0 = *(const v4f*)(bSrc0);
    v4f gb1 = *(const v4f*)(bSrc1);
    *(v4f*)&As[0][ar][4 * ac]      = ga;
    *(v4f*)&Bs[0][br][4 * bc]      = GUARD ? gb0 * sB0 : gb0;
    *(v4f*)&Bs[0][br + 64][4 * bc] = GUARD ? gb1 * sB1 : gb1;
  }
  __syncthreads();

  const int nch = K / KC;
  for (int c = 0; c < nch; ++c) {
    const int p = c & 1;
    v4f na = {}, nb0 = {}, nb1 = {};
    const bool more = (c + 1 < nch);
    if (more) {
      const int k0 = (c + 1) * KC;          // prefetch next chunk to regs
      na  = *(const v4f*)(aSrc  + k0);
      nb0 = *(const v4f*)(bSrc0 + k0);
      nb1 = *(const v4f*)(bSrc1 + k0);
    }
    compute(As[p], Bs[p]);                   // 16 WMMAs fed from LDS
    if (more) {
      *(v4f*)&As[p ^ 1][ar][4 * ac]      = na;
      *(v4f*)&Bs[p ^ 1][br][4 * bc]      = GUARD ? nb0 * sB0 : nb0;
      *(v4f*)&Bs[p ^ 1][br + 64][4 * bc] = GUARD ? nb1 * sB1 : nb1;
      __syncthreads();
    }
  }

  auto epi = [&](v8f acc, int col) {
#pragma unroll
    for (int v = 0; v < 8; ++v) {
      int row = m0 + v + 8 * half;
      float x = acc[v];
      if (GUARD) x += (col < bias_n) ? bias[col] : 0.f;
      else       x += bias[col];
      if (act == 1)      x = fmaxf(x, 0.f);
      else if (act == 2) x = (x > 0.f) ? x : expm1f(x);
      if (resid && (!GUARD || col < N)) x += resid[(size_t)row * ldr + col];
      if (!GUARD || col < N) C[(size_t)row * ldc + col] = x;
    }
  };
  epi(acc0, n0 + l15);
  epi(acc1, n0 + 16 + l15);
  epi(acc2, n0 + 32 + l15);
  epi(acc3, n0 + 48 + l15);
}

// --------------------------------------------------------------------------
// Build x_cat[2048,256] = [ emb(+freq) | mask ]
// emb = mask!=0 ? x*nan_w + nan_b + freq : nan_val + freq,  x = nan_to_num(data,-1)
// --------------------------------------------------------------------------
__global__ void k_xcat(const float* __restrict__ data, const float* __restrict__ mask,
                       const float* __restrict__ nw, const float* __restrict__ nb,
                       const float* __restrict__ freq, const float* __restrict__ nv,
                       float* __restrict__ xcat)
{
  int idx = blockIdx.x * 256 + threadIdx.x;           // 2048*256 total
  int row = idx >> 8, c = idx & 255, s = row & 63;
  float m = mask[row];
  float v;
  if (c < FEAT_) {
    if (m == 0.f) v = nv[c] + freq[s * FEAT_ + c];
    else {
      float x = data[row];
      if (x != x) x = -1.f;                            // nan_to_num(nan=-1)
      v = x * nw[c] + nb[c] + freq[s * FEAT_ + c];
    }
  } else {
    v = m;
  }
  xcat[idx] = v;
}

// Add sinusoidal PE to x0[2048,256]
__global__ void k_add_pe(float* __restrict__ x)
{
  int idx = blockIdx.x * 256 + threadIdx.x;
  int row = idx >> 8, c = idx & 255, s = row & 63;
  float e = -logf(10000.f) * (float)(2 * (c >> 1)) / (float)DM;
  float ang = (float)s * expf(e);
  x[idx] += (c & 1) ? cosf(ang) : sinf(ang);
}

// LayerNorm over last dim (256): one wave32 per row, 8 rows/block.
__global__ __launch_bounds__(256) void k_layernorm(
    const float* __restrict__ x, const float* __restrict__ g,
    const float* __restrict__ b, float* __restrict__ o)
{
  int row = blockIdx.x * 8 + (threadIdx.x >> 5);
  int lane = threadIdx.x & 31;
  const float* xr = x + (size_t)row * DM;
  float vals[8], s = 0.f, s2 = 0.f;
#pragma unroll
  for (int i = 0; i < 8; ++i) {
    float v = xr[lane + 32 * i];
    vals[i] = v; s += v; s2 += v * v;
  }
#pragma unroll
  for (int off = 16; off; off >>= 1) {
    s  += __shfl_xor(s,  off, 32);
    s2 += __shfl_xor(s2, off, 32);
  }
  float mu = s * (1.f / DM);
  float var = s2 * (1.f / DM) - mu * mu;
  float r = rsqrtf(var + 1e-5f);
  float* orow = o + (size_t)row * DM;
#pragma unroll
  for (int i = 0; i < 8; ++i) {
    int c = lane + 32 * i;
    orow[c] = (vals[i] - mu) * r * g[c] + b[c];
  }
}

// Attention per (head, batch) block: 64 threads = 64 query rows.
// qkv row layout [q(256)|k(256)|v(256)]; head h uses cols h*64..h*64+63.
__global__ __launch_bounds__(64) void k_attn(const float* __restrict__ qkv,
                                             float* __restrict__ out)
{
  __shared__ float qs[64][65];
  __shared__ float kv[64][65];
  __shared__ float ss[64][65];
  int h = blockIdx.x, b = blockIdx.y;
  int t = threadIdx.x;
  const float* base = qkv + ((size_t)(b * 64 + t)) * (3 * DM) + h * 64;
  for (int d = 0; d < 64; ++d) { qs[t][d] = base[d]; kv[t][d] = base[DM + d]; }
  __syncthreads();
  float mx = -INFINITY;
  for (int j = 0; j < 64; ++j) {
    float acc = 0.f;
    for (int d = 0; d < 64; ++d) acc += qs[t][d] * kv[j][d];
    float sc = (j == t) ? -INFINITY : acc * 0.125f;  // 1/sqrt(64), diag masked
    ss[t][j] = sc; mx = fmaxf(mx, sc);
  }
  float den = 0.f;
  for (int j = 0; j < 64; ++j) { float e = expf(ss[t][j] - mx); ss[t][j] = e; den += e; }
  float inv = 1.f / den;
  __syncthreads();
  for (int d = 0; d < 64; ++d) kv[t][d] = base[2 * DM + d];   // reload as V
  __syncthreads();
  float* orow = out + ((size_t)(b * 64 + t)) * DM + h * 64;
  for (int d = 0; d < 64; ++d) {
    float acc = 0.f;
    for (int j = 0; j < 64; ++j) acc += ss[t][j] * kv[j][d];
    orow[d] = acc * inv;
  }
}

// Block-wide sum reduction helper (256 threads)
__device__ float blockReduceSum(float x, float* red)
{
  int t = threadIdx.x;
  red[t] = x; __syncthreads();
  for (int s = 128; s > 0; s >>= 1) {
    if (t < s) red[t] += red[t + s];
    __syncthreads();
  }
  float r = red[0]; __syncthreads();
  return r;
}

// Spectral norm: 15 power iterations, then write w/sigma zero-padded to
// out[outR, outC]. One block of 256 threads per matrix.
__global__ __launch_bounds__(256) void k_specnorm(
    const float* __restrict__ w, int dout, int din,
    float* __restrict__ out, int outR, int outC)
{
  __shared__ float u[512], v[512], red[256];
  int t = threadIdx.x;
  for (int i = t; i < dout; i += 256) u[i] = rsqrtf((float)dout);
  __syncthreads();
  for (int it = 0; it < 15; ++it) {
    for (int j = t; j < din; j += 256) {
      float a = 0.f;
      for (int i = 0; i < dout; ++i) a += w[(size_t)i * din + j] * u[i];
      v[j] = a;
    }
    __syncthreads();
    float s = 0.f;
    for (int j = t; j < din; j += 256) s += v[j] * v[j];
    s = blockReduceSum(s, red);
    float inv = 1.f / (sqrtf(s) + 1e-12f);
    for (int j = t; j < din; j += 256) v[j] *= inv;
    __syncthreads();
    for (int i = t; i < dout; i += 256) {
      float a = 0.f;
      for (int j = 0; j < din; ++j) a += w[(size_t)i * din + j] * v[j];
      u[i] = a;
    }
    __syncthreads();
    float su = 0.f;
    for (int i = t; i < dout; i += 256) su += u[i] * u[i];
    su = blockReduceSum(su, red);
    float invu = 1.f / (sqrtf(su) + 1e-12f);
    for (int i = t; i < dout; i += 256) u[i] *= invu;
    __syncthreads();
  }
  // sigma = u . (W v)
  float sg = 0.f;
  for (int i = t; i < dout; i += 256) {
    float a = 0.f;
    for (int j = 0; j < din; ++j) a += w[(size_t)i * din + j] * v[j];
    sg += u[i] * a;
  }
  sg = blockReduceSum(sg, red);
  float invs = 1.f / sg;
  for (int idx = t; idx < outR * outC; idx += 256) {
    int i = idx / outC, j = idx % outC;
    out[idx] = (i < dout && j < din) ? w[(size_t)i * din + j] * invs : 0.f;
  }
}

// dt = max(tsat)/NUM_STEPS   (B=32 values, one wave)
__global__ void k_dt(const float* __restrict__ tsat, float* __restrict__ dt)
{
  float v = tsat[threadIdx.x];
#pragma unroll
  for (int off = 16; off; off >>= 1) v = fmaxf(v, __shfl_xor(v, off, 32));
  if (threadIdx.x == 0) dt[0] = v * (1.f / 8.f);
}

// Initialize z[2048,272] = [y0(6) | B1 | tsat | offsets | enc_x(256) | 0-pad(7)]
// and y[2048,6] = h0  (cols 2 and 5 = 1).
__global__ void k_init_z(float* __restrict__ z, float* __restrict__ y,
                         const float* __restrict__ B1, const float* __restrict__ tsat,
                         const float* __restrict__ offs, const float* __restrict__ encx)
{
  int idx = blockIdx.x * 256 + threadIdx.x;           // 2048*272 total
  int row = idx / KODE, c = idx % KODE;
  if (row >= MTOT) return;
  int b = row >> 6;
  float v;
  if (c < 6) {
    v = (c == 2 || c == 5) ? 1.f : 0.f;
    y[row * 6 + c] = v;
  } else if (c == 6)  v = B1[b];
  else if (c == 7)    v = tsat[b];
  else if (c == 8)    v = offs[row];
  else if (c < 265)   v = encx[(size_t)row * DM + (c - 9)];
  else                v = 0.f;
  z[idx] = v;
}

// z[:,0:6] = y + dt * sum(ci * ki);  optionally also update y (final stage).
// k buffers are [2048,16] (cols >= 6 are zero).
__global__ void k_compose(float* __restrict__ z, float* __restrict__ y,
                          const float* __restrict__ dtp,
                          const float* __restrict__ k1, const float* __restrict__ k2,
                          const float* __restrict__ k3, const float* __restrict__ k4,
                          const float* __restrict__ k5, const float* __restrict__ k6,
                          float c1, float c2, float c3, float c4, float c5, float c6,
                          int nk, int write_y)
{
  int idx = blockIdx.x * 256 + threadIdx.x;           // 2048*6 total
  if (idx >= MTOT * 6) return;
  int row = idx / 6, c = idx % 6;
  int o = row * 16 + c;
  float acc = c1 * k1[o];
  if (nk > 1) acc += c2 * k2[o];
  if (nk > 2) acc += c3 * k3[o];
  if (nk > 3) acc += c4 * k4[o];
  if (nk > 4) acc += c5 * k5[o];
  if (nk > 5) acc += c6 * k6[o];
  float v = y[idx] + dtp[0] * acc;
  z[row * KODE + c] = v;
  if (write_y) y[idx] = v;
}

__global__ void k_copy(const float* __restrict__ src, float* __restrict__ dst, int n)
{
  int i = blockIdx.x * 256 + threadIdx.x;
  if (i < n) dst[i] = src[i];
}

// ---------------------------------------------------------------------------
extern "C" void kernel_launch(void* const* d_in, const int* in_sizes, int n_in,
                              void* d_out, int out_size, void* d_ws, size_t ws_size,
                              hipStream_t stream)
{
  (void)in_sizes; (void)n_in; (void)out_size; (void)ws_size;
  const float* data  = (const float*)d_in[0];
  const float* B1p   = (const float*)d_in[1];
  const float* tsat  = (const float*)d_in[2];
  const float* mask  = (const float*)d_in[3];
  const float* offs  = (const float*)d_in[4];
  const float* nan_w = (const float*)d_in[5];
  const float* nan_b = (const float*)d_in[6];
  const float* freq  = (const float*)d_in[7];
  const float* nanv  = (const float*)d_in[8];
  const float* enc_w = (const float*)d_in[9];
  const float* enc_b = (const float*)d_in[10];
  const float* ln1g  = (const float*)d_in[11];
  const float* ln1b  = (const float*)d_in[12];
  const float* aiw   = (const float*)d_in[13];
  const float* aib   = (const float*)d_in[14];
  const float* aow   = (const float*)d_in[15];
  const float* aob   = (const float*)d_in[16];
  const float* flg   = (const float*)d_in[17];
  const float* flb   = (const float*)d_in[18];
  const float* fw1   = (const float*)d_in[19];
  const float* fb1   = (const float*)d_in[20];
  const float* fw2   = (const float*)d_in[21];
  const float* fb2   = (const float*)d_in[22];
  const float* ow[5]; const float* ob[5];
  for (int i = 0; i < 5; ++i) {
    ow[i] = (const float*)d_in[23 + 2 * i];
    ob[i] = (const float*)d_in[24 + 2 * i];
  }

  // Workspace layout (floats)
  float* F = (float*)d_ws;
  float* x0    = F; F += MTOT * DM;
  float* xn    = F; F += MTOT * DM;
  float* qkv   = F; F += MTOT * 3 * DM;
  float* x1    = F; F += MTOT * DM;
  float* ffnh  = F; F += MTOT * INNER_;
  float* encx  = F; F += MTOT * DM;
  float* xcat  = F; F += MTOT * DM;          // reused as attention output
  float* wh0   = F; F += HID_ * KODE;
  float* wh1   = F; F += HID_ * HID_;
  float* wh2   = F; F += HID_ * HID_;
  float* wh3   = F; F += HID_ * HID_;
  float* wh4   = F; F += 16 * HID_;
  float* z     = F; F += MTOT * KODE;
  float* ha    = F; F += MTOT * HID_;
  float* hb    = F; F += MTOT * HID_;
  float* kbuf[6];
  for (int i = 0; i < 6; ++i) { kbuf[i] = F; F += MTOT * 16; }
  float* y     = F; F += MTOT * 6;
  float* dtp   = F; F += 1;

  auto gemm = [&](const float* A, int lda, const float* W, int ldw,
                  const float* bias, int bias_n, float* C, int ldc,
                  const float* resid, int ldr, int N, int K, int act) {
    dim3 g(cdiv(N, 128), MTOT / 64);
    if (N % 128 == 0)
      k_gemm<0><<<g, 256, 0, stream>>>(A, lda, W, ldw, bias, bias_n, C, ldc,
                                       resid, ldr, N, K, act);
    else
      k_gemm<1><<<g, 256, 0, stream>>>(A, lda, W, ldw, bias, bias_n, C, ldc,
                                       resid, ldr, N, K, act);
  };

  // --- embedding + encoder + PE ---
  k_xcat<<<MTOT, 256, 0, stream>>>(data, mask, nan_w, nan_b, freq, nanv, xcat);
  gemm(xcat, DM, enc_w, DM, enc_b, DM, x0, DM, nullptr, 0, DM, DM, 0);
  k_add_pe<<<MTOT, 256, 0, stream>>>(x0);

  // --- attention block ---
  k_layernorm<<<MTOT / 8, 256, 0, stream>>>(x0, ln1g, ln1b, xn);
  gemm(xn, DM, aiw, DM, aib, 3 * DM, qkv, 3 * DM, nullptr, 0, 3 * DM, DM, 0);
  k_attn<<<dim3(4, B_), 64, 0, stream>>>(qkv, xcat);   // xcat := attn output
  gemm(xcat, DM, aow, DM, aob, DM, x1, DM, x0, DM, DM, DM, 0);

  // --- FFN block ---
  k_layernorm<<<MTOT / 8, 256, 0, stream>>>(x1, flg, flb, xn);
  gemm(xn, DM, fw1, DM, fb1, INNER_, ffnh, INNER_, nullptr, 0, INNER_, DM, 1);
  gemm(ffnh, INNER_, fw2, INNER_, fb2, DM, encx, DM, x1, DM, DM, INNER_, 0);

  // --- spectral norms (padded) ---
  k_specnorm<<<1, 256, 0, stream>>>(ow[0], HID_, 265, wh0, HID_, KODE);
  k_specnorm<<<1, 256, 0, stream>>>(ow[1], HID_, HID_, wh1, HID_, HID_);
  k_specnorm<<<1, 256, 0, stream>>>(ow[2], HID_, HID_, wh2, HID_, HID_);
  k_specnorm<<<1, 256, 0, stream>>>(ow[3], HID_, HID_, wh3, HID_, HID_);
  k_specnorm<<<1, 256, 0, stream>>>(ow[4], SDIM_, HID_, wh4, 16, HID_);

  // --- ODE setup ---
  k_dt<<<1, 32, 0, stream>>>(tsat, dtp);
  k_init_z<<<cdiv(MTOT * KODE, 256), 256, 0, stream>>>(z, y, B1p, tsat, offs, encx);

  auto fchain = [&](float* kout) {
    gemm(z,  KODE, wh0, KODE, ob[0], HID_, ha, HID_, nullptr, 0, HID_, KODE, 2);
    gemm(ha, HID_, wh1, HID_, ob[1], HID_, hb, HID_, nullptr, 0, HID_, HID_, 2);
    gemm(hb, HID_, wh2, HID_, ob[2], HID_, ha, HID_, nullptr, 0, HID_, HID_, 2);
    gemm(ha, HID_, wh3, HID_, ob[3], HID_, hb, HID_, nullptr, 0, HID_, HID_, 2);
    gemm(hb, HID_, wh4, HID_, ob[4], SDIM_, kout, 16, nullptr, 0, 16, HID_, 0);
  };
  auto compose = [&](float c1, float c2, float c3, float c4, float c5, float c6,
                     int nk, int wy) {
    k_compose<<<cdiv(MTOT * 6, 256), 256, 0, stream>>>(
        z, y, dtp, kbuf[0], kbuf[1], kbuf[2], kbuf[3], kbuf[4], kbuf[5],
        c1, c2, c3, c4, c5, c6, nk, wy);
  };

  // --- 8 fixed dopri5 steps ---
  for (int step = 0; step < 8; ++step) {
    fchain(kbuf[0]);                                   // k1 = f(y)
    compose(0.2f, 0, 0, 0, 0, 0, 1, 0);
    fchain(kbuf[1]);                                   // k2
    compose(3.f/40.f, 9.f/40.f, 0, 0, 0, 0, 2, 0);
    fchain(kbuf[2]);                                   // k3
    compose(44.f/45.f, -56.f/15.f, 32.f/9.f, 0, 0, 0, 3, 0);
    fchain(kbuf[3]);                                   // k4
    compose((float)(19372.0/6561.0), (float)(-25360.0/2187.0),
            (float)(64448.0/6561.0), (float)(-212.0/729.0), 0, 0, 4, 0);
    fchain(kbuf[4]);                                   // k5
    compose((float)(9017.0/3168.0), (float)(-355.0/33.0),
            (float)(46732.0/5247.0), (float)(49.0/176.0),
            (float)(-5103.0/18656.0), 0, 5, 0);
    fchain(kbuf[5]);                                   // k6
    compose((float)(35.0/384.0), 0.f, (float)(500.0/1113.0),
            (float)(125.0/192.0), (float)(-2187.0/6784.0),
            (float)(11.0/84.0), 6, 1);                 // y += ...; z := new y
  }

  k_copy<<<cdiv(MTOT * 6, 256), 256, 0, stream>>>(y, (float*)d_out, MTOT * 6);
}